// NGCF_10514079940677
// MI455X (gfx1250) — compile-verified
//
#include <hip/hip_runtime.h>
#include <hip/hip_bf16.h>

// ---------------------------------------------------------------------------
// NGCF forward on MI455X (gfx1250, wave32, WMMA).
//
// side = spmm(adj, ego)                 -> atomics into L2-resident buffer
// gc   = leaky_relu(side @ Wg + bg)     -> V_WMMA_F32_16X16X4_F32
// bi   = leaky_relu((ego*side) @ Wb+bb) -> V_WMMA_F32_16X16X4_F32
// ego  = gc + bi ; out block = l2norm(ego)
// ---------------------------------------------------------------------------

#define N_USERS 50000
#define N_ITEMS 100000
#define N_TOTAL (N_USERS + N_ITEMS)   // 150000
#define EMB 128
#define OUT_W 512                     // EMB * (N_LAYERS + 1)
#define N_LAYERS 3

typedef __attribute__((ext_vector_type(2))) float v2f;
typedef __attribute__((ext_vector_type(8))) float v8f;

// ---------------------------------------------------------------------------
// ego = concat(user_emb, item_emb); out[:, 0:128] = ego  (stage-0 block)
// ---------------------------------------------------------------------------
__global__ __launch_bounds__(256) void ngcf_init_kernel(
    const float* __restrict__ user_emb, const float* __restrict__ item_emb,
    float* __restrict__ ego, float* __restrict__ out)
{
    size_t idx = (size_t)blockIdx.x * blockDim.x + threadIdx.x;
    if (idx >= (size_t)N_TOTAL * EMB) return;
    int row = (int)(idx >> 7);
    int col = (int)(idx & 127);
    float v = (row < N_USERS) ? user_emb[(size_t)row * EMB + col]
                              : item_emb[(size_t)(row - N_USERS) * EMB + col];
    ego[idx] = v;
    out[(size_t)row * OUT_W + col] = v;
}

__global__ __launch_bounds__(256) void ngcf_zero_kernel(float* __restrict__ p)
{
    size_t idx = (size_t)blockIdx.x * blockDim.x + threadIdx.x;
    if (idx < (size_t)N_TOTAL * EMB) p[idx] = 0.0f;
}

// ---------------------------------------------------------------------------
// SpMM: one wave32 per nonzero. Each lane moves a float4 of the 128-wide row
// (32 lanes x 16B = 512B) and scatter-adds into side with f32 atomics.
// ego/side are 76.8MB each -> L2 (192MB) resident: gathers & atomics hit L2.
// ---------------------------------------------------------------------------
__global__ __launch_bounds__(256) void ngcf_spmm_kernel(
    const int* __restrict__ rows, const int* __restrict__ cols,
    const float* __restrict__ vals, const float* __restrict__ ego,
    float* __restrict__ side, int nnz)
{
    int gtid = blockIdx.x * blockDim.x + threadIdx.x;
    int edge = gtid >> 5;
    int lane = gtid & 31;
    if (edge >= nnz) return;
    int r = rows[edge];
    int c = cols[edge];
    float w = vals[edge];
    const float4* src = (const float4*)(ego + (size_t)c * EMB);
    float4 x = src[lane];
    float* dst = side + (size_t)r * EMB + lane * 4;
    atomicAdd(dst + 0, x.x * w);
    atomicAdd(dst + 1, x.y * w);
    atomicAdd(dst + 2, x.z * w);
    atomicAdd(dst + 3, x.w * w);
}

// ---------------------------------------------------------------------------
// Fused dual-GEMM + bias + leaky_relu + add + l2norm, full f32 via
// V_WMMA_F32_16X16X4_F32. 4 waves/block, one 16-row tile per wave.
// Weight column panels (128x16 per matrix) staged in LDS per block.
//
// f32 16x16x4 WMMA VGPR layout (ISA 7.12.2):
//   A (16x4): lane L<16 -> row L, {v0:K=4k, v1:K=4k+1};  L>=16 -> {K=4k+2,4k+3}
//   B (4x16): lane L<16 -> col L, {v0:K=4k, v1:K=4k+1};  L>=16 -> {K=4k+2,4k+3}
//   C (16x16): vgpr v, lane L -> M = v + (L>=16 ? 8:0), N = L&15
// ---------------------------------------------------------------------------
__global__ __launch_bounds__(128) void ngcf_layer_kernel(
    const float* __restrict__ side, float* __restrict__ ego,
    const float* __restrict__ gcw, const float* __restrict__ gcb,
    const float* __restrict__ biw, const float* __restrict__ bib,
    float* __restrict__ out, int layer_off, int num_tiles)
{
    __shared__ float sGC[EMB * 16];   // gc_w panel  [k][n], n in 0..15
    __shared__ float sBI[EMB * 16];   // bi_w panel

    const int tid  = threadIdx.x;
    const int wave = tid >> 5;
    const int lane = tid & 31;
    const int hf   = lane >> 4;       // 0: K pair {0,1}; 1: K pair {2,3}
    const int l    = lane & 15;
    const int tile = blockIdx.x * 4 + wave;
    const bool active = tile < num_tiles;   // uniform per wave -> EXEC all-1s

    const int R    = tile * 16;
    const int kOff = hf * 2;
    const float* sideRow = side + (size_t)(R + l) * EMB;
    const float* egoRow  = ego  + (size_t)(R + l) * EMB;

    float egoNew[64];                 // [j][v]: 8 col panels x 8 C-vgprs

    for (int j = 0; j < 8; ++j) {
        __syncthreads();              // previous panel fully consumed
        // Stage both 128x16 weight panels (coalesced, all 128 threads).
        for (int idx = tid; idx < EMB * 16; idx += 128) {
            int kk = idx >> 4, n = idx & 15;
            sGC[idx] = gcw[(size_t)kk * EMB + j * 16 + n];
            sBI[idx] = biw[(size_t)kk * EMB + j * 16 + n];
        }
        __syncthreads();

        if (active) {
            v8f accG = {};
            v8f accB = {};
#pragma unroll
            for (int k = 0; k < 32; ++k) {
                const int kb = 4 * k + kOff;
                v2f aS = *(const v2f*)(sideRow + kb);   // side fragment
                v2f aE = *(const v2f*)(egoRow  + kb);   // ego fragment
                v2f aM = aS * aE;                       // (ego*side) fragment
                v2f bG, bB;
                bG.x = sGC[(kb + 0) * 16 + l];
                bG.y = sGC[(kb + 1) * 16 + l];
                bB.x = sBI[(kb + 0) * 16 + l];
                bB.y = sBI[(kb + 1) * 16 + l];
                accG = __builtin_amdgcn_wmma_f32_16x16x4_f32(
                    false, aS, false, bG, (short)0, accG, false, false);
                accB = __builtin_amdgcn_wmma_f32_16x16x4_f32(
                    false, aM, false, bB, (short)0, accB, false, false);
            }
            const float bg = gcb[j * 16 + l];
            const float bb = bib[j * 16 + l];
#pragma unroll
            for (int v = 0; v < 8; ++v) {
                float g = accG[v] + bg; g = (g >= 0.0f) ? g : 0.01f * g;
                float b = accB[v] + bb; b = (b >= 0.0f) ? b : 0.01f * b;
                egoNew[j * 8 + v] = g + b;
            }
        }
    }

    if (!active) return;

    // Row-wise l2 norm: lane holds rows {v + hf*8}; reduce squares over the
    // 16 lanes of each half (xor masks 1/2/4/8 stay within the half).
    float p[8];
#pragma unroll
    for (int v = 0; v < 8; ++v) {
        float s = 0.0f;
#pragma unroll
        for (int j = 0; j < 8; ++j) { float e = egoNew[j * 8 + v]; s += e * e; }
        p[v] = s;
    }
#pragma unroll
    for (int mask = 1; mask <= 8; mask <<= 1) {
#pragma unroll
        for (int v = 0; v < 8; ++v) p[v] += __shfl_xor(p[v], mask, 32);
    }
    float inv[8];
#pragma unroll
    for (int v = 0; v < 8; ++v)
        inv[v] = 1.0f / fmaxf(sqrtf(p[v]), 1e-12f);

    // Write un-normalized ego (carried to next layer) + normalized out block.
#pragma unroll
    for (int j = 0; j < 8; ++j) {
#pragma unroll
        for (int v = 0; v < 8; ++v) {
            int row = R + v + hf * 8;
            int col = j * 16 + l;
            float e = egoNew[j * 8 + v];
            ego[(size_t)row * EMB + col] = e;
            out[(size_t)row * OUT_W + layer_off + col] = e * inv[v];
        }
    }
}

// ---------------------------------------------------------------------------
extern "C" void kernel_launch(void* const* d_in, const int* in_sizes, int n_in,
                              void* d_out, int out_size, void* d_ws, size_t ws_size,
                              hipStream_t stream) {
    const float* user_emb = (const float*)d_in[0];
    const float* item_emb = (const float*)d_in[1];
    const float* gc_w     = (const float*)d_in[2];   // [3,128,128]
    const float* gc_b     = (const float*)d_in[3];   // [3,128]
    const float* bi_w     = (const float*)d_in[4];   // [3,128,128]
    const float* bi_b     = (const float*)d_in[5];   // [3,128]
    const int*   adj_rows = (const int*)d_in[6];
    const int*   adj_cols = (const int*)d_in[7];
    const float* adj_vals = (const float*)d_in[8];
    const int nnz = in_sizes[6];

    float* out  = (float*)d_out;
    float* ego  = (float*)d_ws;                        // N*128 f32
    float* side = ego + (size_t)N_TOTAL * EMB;         // N*128 f32

    const size_t elems = (size_t)N_TOTAL * EMB;
    const int eb = (int)((elems + 255) / 256);
    const int num_tiles = N_TOTAL / 16;                // 9375
    const int layer_blocks = (num_tiles + 3) / 4;
    const int spmm_blocks = (int)(((size_t)nnz * 32 + 255) / 256);

    ngcf_init_kernel<<<eb, 256, 0, stream>>>(user_emb, item_emb, ego, out);

    for (int L = 0; L < N_LAYERS; ++L) {
        ngcf_zero_kernel<<<eb, 256, 0, stream>>>(side);
        ngcf_spmm_kernel<<<spmm_blocks, 256, 0, stream>>>(
            adj_rows, adj_cols, adj_vals, ego, side, nnz);
        ngcf_layer_kernel<<<layer_blocks, 128, 0, stream>>>(
            side, ego,
            gc_w + (size_t)L * EMB * EMB, gc_b + (size_t)L * EMB,
            bi_w + (size_t)L * EMB * EMB, bi_b + (size_t)L * EMB,
            out, (L + 1) * EMB, num_tiles);
    }
}